// Patch5Model_80573586473196
// MI455X (gfx1250) — compile-verified
//
#include <hip/hip_runtime.h>
#include <hip/hip_bf16.h>
#include <math.h>

typedef __attribute__((ext_vector_type(2))) float v2f;
typedef __attribute__((ext_vector_type(8))) float v8f;

#define B_   64
#define C_   128
#define H_   56
#define HW_  (H_ * H_)        // 3136, divisible by 16

// ---------------------------------------------------------------------------
// Kernel 1: channel-sum S[b,h,w] = sum_c fm[b,c,h,w] via V_WMMA_F32_16X16X4_F32
//   D(16x16) = A(16x4, all ones) x B(4x16 = 4 channels x 16 positions) + C
// 32 WMMAs accumulate the 128-channel sum for 16 contiguous positions / wave.
// Memory-bound (102.8 MB streamed once); WMMA redundancy (16 equal rows) is free.
// A is all-ones, so the result only depends on B's "column = lane%16" striping.
// ---------------------------------------------------------------------------
__global__ __launch_bounds__(256) void chansum_wmma(const float* __restrict__ fm,
                                                    float* __restrict__ S) {
  const int lane = threadIdx.x & 31;
  const int wave = threadIdx.x >> 5;
  const int gw   = blockIdx.x * 8 + wave;   // global wave id
  const int posBase = gw * 16;              // 16 positions per wave (stays in one batch)
  const int b   = posBase / HW_;
  const int pos = posBase % HW_;
  const int n   = lane & 15;                // column (position) index
  const int hf  = lane >> 4;                // half-wave selects K slots {0,1} vs {2,3}... (any perm ok)

  // lane's element stream: channel base offset 2*hf, position pos+n
  const float* p = fm + ((size_t)b * C_ + (size_t)(2 * hf)) * HW_ + pos + n;

  v2f a; a[0] = 1.0f; a[1] = 1.0f;          // A = ones(16x4): every slot 1.0
  v8f acc = {};
  #pragma unroll 4
  for (int c0 = 0; c0 < C_; c0 += 4) {
    v2f bv;
    bv[0] = p[(size_t)c0 * HW_];            // channel c0 + 2*hf
    bv[1] = p[(size_t)c0 * HW_ + HW_];      // channel c0 + 2*hf + 1
    acc = __builtin_amdgcn_wmma_f32_16x16x4_f32(
        /*neg_a=*/false, a, /*neg_b=*/false, bv,
        /*c_mod=*/(short)0, acc, /*reuse_a=*/false, /*reuse_b=*/false);
  }
  // D layout: VGPR0, lane L -> D[M = (L<16?0:8), N = L%16]; all rows equal.
  if (lane < 16) S[(size_t)b * HW_ + pos + n] = acc[0];
}

// ---------------------------------------------------------------------------
// Kernel 2: per (batch, type): avg-pool, then 3x (argmax -> box math -> erase)
// One 256-thread block per (b, type); everything LDS-resident.
// ---------------------------------------------------------------------------
__global__ __launch_bounds__(256) void peaks_kernel(const float* __restrict__ S,
                                                    const int* __restrict__ scale,
                                                    float* __restrict__ out) {
  __shared__ float sS[HW_];     // channel-summed plane
  __shared__ float sc[3025];    // pooled score map (max 55x55)
  __shared__ float rV[256];
  __shared__ int   rI[256];

  const int tid  = threadIdx.x;
  const int type = blockIdx.x & 1;
  const int b    = blockIdx.x >> 1;

  for (int i = tid; i < HW_; i += 256) sS[i] = S[(size_t)b * HW_ + i];
  __syncthreads();

  const int   k   = (type == 0) ? 3 : 2;
  const int   Hs  = H_ - k + 1;             // 54 or 55
  const int   N   = Hs * Hs;
  const float inv = 1.0f / (float)(k * k);
  for (int idx = tid; idx < N; idx += 256) {
    int i = idx / Hs, j = idx % Hs;
    float s = 0.0f;
    for (int di = 0; di < k; ++di)
      for (int dj = 0; dj < k; ++dj)
        s += sS[(i + di) * H_ + (j + dj)];
    sc[idx] = s * inv;                      // == sum_c avg_pool_k(fm_c)
  }
  __syncthreads();

  const int s0   = scale[b * 2 + 0];
  const int s1   = scale[b * 2 + 1];
  const int smin = (s0 < s1) ? s0 : s1;
  const int base0 = (s0 - smin) >> 1;       // nonneg -> same as //2
  const int base1 = (s1 - smin) >> 1;
  const int ph    = (type == 0) ? 224 : 112;
  const int hp    = ph >> 1;                // ph even -> ph%2 == 0

  float* locO = out;                        // [64][6][4]
  float* fpsO = out + B_ * 6 * 4;           // [64][6][2]
  float* valO = out + B_ * 6 * 4 + B_ * 6 * 2; // [64][6]

  for (int iter = 0; iter < 3; ++iter) {
    // ---- block argmax, first-index tie-break (matches jnp.argmax) ----
    float bv = -INFINITY; int bi = 0;
    for (int idx = tid; idx < N; idx += 256) {
      float v = sc[idx];
      if (v > bv) { bv = v; bi = idx; }     // ascending scan: strict > keeps lowest idx
    }
    rV[tid] = bv; rI[tid] = bi;
    __syncthreads();
    for (int off = 128; off > 0; off >>= 1) {
      if (tid < off) {
        float ov = rV[tid + off]; int oi = rI[tid + off];
        if (ov > rV[tid] || (ov == rV[tid] && oi < rI[tid])) { rV[tid] = ov; rI[tid] = oi; }
      }
      __syncthreads();
    }
    const float vmax = rV[0];
    const int   pidx = rI[0];

    // ---- outputs (thread 0) ----
    if (tid == 0) {
      int lr = pidx / Hs, lc = pidx % Hs;
      float rate_h = (float)(2 * lr + H_ - Hs + 1) / (2.0f * (float)H_);
      float rate_w = (float)(2 * lc + H_ - Hs + 1) / (2.0f * (float)H_);
      int c0 = (int)((float)base0 + (float)smin * rate_h);   // astype(int32): truncate
      int c1 = (int)((float)base1 + (float)smin * rate_w);
      int top = c0 - hp, bot = c0 + hp;
      int lef = c1 - hp, rig = c1 + hp;
      int below0 = top < 0 ? top : 0;  int below1 = lef < 0 ? lef : 0;
      top -= below0; bot -= below0; lef -= below1; rig -= below1;
      int over0 = bot - s0 > 0 ? bot - s0 : 0;
      int over1 = rig - s1 > 0 ? rig - s1 : 0;
      top = (top - over0 > 0) ? top - over0 : 0;
      lef = (lef - over1 > 0) ? lef - over1 : 0;
      bot -= over0; rig -= over1;
      int e = type * 3 + iter;
      locO[(b * 6 + e) * 4 + 0] = (float)top;
      locO[(b * 6 + e) * 4 + 1] = (float)lef;
      locO[(b * 6 + e) * 4 + 2] = (float)bot;
      locO[(b * 6 + e) * 4 + 3] = (float)rig;
      fpsO[(b * 6 + e) * 2 + 0] = (float)lr;
      fpsO[(b * 6 + e) * 2 + 1] = (float)lc;
      valO[b * 6 + e] = vmax;
    }

    // ---- NMS erase: zero positions whose 3x3 (clipped) neighborhood holds vmax ----
    // Two-pass: flag against pre-erase values, then apply (matches max_pool==vmax mask).
    unsigned int flags = 0u;
    int slot = 0;
    for (int idx = tid; idx < N; idx += 256, ++slot) {
      int i = idx / Hs, j = idx % Hs;
      bool hit = false;
      for (int di = -1; di <= 1; ++di) {
        int ii = i + di; if (ii < 0 || ii >= Hs) continue;
        for (int dj = -1; dj <= 1; ++dj) {
          int jj = j + dj; if (jj < 0 || jj >= Hs) continue;
          if (sc[ii * Hs + jj] == vmax) hit = true;
        }
      }
      if (hit) flags |= (1u << slot);
    }
    __syncthreads();
    slot = 0;
    for (int idx = tid; idx < N; idx += 256, ++slot)
      if (flags & (1u << slot)) sc[idx] = 0.0f;   // reference multiplies by 0 mask
    __syncthreads();
  }
}

// ---------------------------------------------------------------------------
extern "C" void kernel_launch(void* const* d_in, const int* in_sizes, int n_in,
                              void* d_out, int out_size, void* d_ws, size_t ws_size,
                              hipStream_t stream) {
  const float* fm    = (const float*)d_in[0];   // [64,128,56,56] f32
  const int*   scale = (const int*)d_in[1];     // [64,2] i32
  float*       out   = (float*)d_out;           // 64*6*4 + 64*6*2 + 64*6 = 2688 floats
  float*       S     = (float*)d_ws;            // [64,3136] f32 scratch (802816 B)

  const int totalWaves = (B_ * HW_) / 16;       // 12544
  chansum_wmma<<<totalWaves / 8, 256, 0, stream>>>(fm, S);
  peaks_kernel<<<B_ * 2, 256, 0, stream>>>(S, scale, out);
}